// LowHighFreqVolume_41721312313574
// MI455X (gfx1250) — compile-verified
//
#include <hip/hip_runtime.h>
#include <hip/hip_bf16.h>

#define B_   2
#define C_   320
#define H_   64
#define W_   128
#define D_   48
#define G_   40
#define CPG  8
#define CR   20   // C / reduction ratio

typedef __attribute__((ext_vector_type(2))) float v2f;
typedef __attribute__((ext_vector_type(8))) float v8f;
typedef int v4i __attribute__((vector_size(16)));   // matches builtin param type

#define AS1 __attribute__((address_space(1)))
#define AS3 __attribute__((address_space(3)))

// generic -> LDS(3) / global(1) pointer casts via integer round-trip
__device__ __forceinline__ AS3 v4i* lds_cast(void* p) {
  return (AS3 v4i*)(unsigned int)(unsigned long long)p;
}
__device__ __forceinline__ AS1 v4i* glob_cast(const void* p) {
  return (AS1 v4i*)(unsigned long long)p;
}

// ---------------- Stage 1: global average pool per (img, b, c) ----------------
__global__ void gap_kernel(const float* __restrict__ ref, const float* __restrict__ tgt,
                           float* __restrict__ gap) {
  __shared__ float red[256];
  const int blk = blockIdx.x;                        // img*(B*C) + b*C + c
  const float* src = (blk < B_ * C_) ? ref : tgt;
  const int off = (blk % (B_ * C_)) * (H_ * W_);
  float s = 0.f;
  for (int i = threadIdx.x; i < H_ * W_; i += 256) s += src[off + i];
  red[threadIdx.x] = s;
  __syncthreads();
  for (int st = 128; st > 0; st >>= 1) {
    if (threadIdx.x < st) red[threadIdx.x] += red[threadIdx.x + st];
    __syncthreads();
  }
  if (threadIdx.x == 0) gap[blk] = red[0] * (1.f / (H_ * W_));
}

// ---------------- Stage 2: FC -> ReLU -> FC -> sigmoid (per img,b) ----------------
__global__ void mask_kernel(const float* __restrict__ gap,
                            const float* __restrict__ w1, const float* __restrict__ b1,
                            const float* __restrict__ w2, const float* __restrict__ b2,
                            float* __restrict__ mask) {
  __shared__ float sg[C_];
  __shared__ float sh[CR];
  const int base = blockIdx.x * C_;                  // (img*B + b) * C
  for (int c = threadIdx.x; c < C_; c += blockDim.x) sg[c] = gap[base + c];
  __syncthreads();
  if (threadIdx.x < CR) {
    float acc = b1[threadIdx.x];
    const float* wr = w1 + threadIdx.x * C_;
    for (int c = 0; c < C_; ++c) acc += wr[c] * sg[c];
    sh[threadIdx.x] = acc > 0.f ? acc : 0.f;
  }
  __syncthreads();
  for (int c = threadIdx.x; c < C_; c += blockDim.x) {
    float acc = b2[c];
    const float* wr = w2 + c * CR;
    for (int j = 0; j < CR; ++j) acc += wr[j] * sh[j];
    mask[base + c] = 1.f / (1.f + __expf(-acc));
  }
}

// ---------------- Stage 3a: zero the w < d padding triangle ----------------
__global__ void ztri_kernel(float* __restrict__ out) {
  const int blk = blockIdx.x;                        // (b*80 + q)*48 + d
  const int d = blk % D_;
  if (d == 0) return;
  const long base = (long)blk * (H_ * W_);
  const int n = d * H_;
  for (int idx = threadIdx.x; idx < n; idx += blockDim.x) {
    const int h = idx / d;
    const int w = idx - h * d;
    out[base + h * W_ + w] = 0.f;
  }
}

// ---------------- Stage 3b: banded groupwise correlation via WMMA ----------------
// one wave per (b, variant, g, h); 4 waves / block; async DMA staging to LDS
__global__ __launch_bounds__(128) void gwc_wmma_kernel(
    const float* __restrict__ ref, const float* __restrict__ tgt,
    const float* __restrict__ mask, float* __restrict__ out) {
  __shared__ float Rl[4][CPG][W_];
  __shared__ float Tl[4][CPG][W_];

  const int lane = threadIdx.x & 31;
  const int wv   = threadIdx.x >> 5;
  const int wid  = blockIdx.x * 4 + wv;
  const int g       = wid % G_;
  const int h       = (wid / G_) % H_;
  const int variant = (wid / (G_ * H_)) & 1;         // 0 = hf (m), 1 = ctr (1-m)
  const int b       = wid / (G_ * H_ * 2);

  // --- async global -> LDS DMA of raw 8x128 panels (no VGPR round-trip) ---
  #pragma unroll
  for (int c = 0; c < CPG; ++c) {
    const long gofs = ((long)(b * C_ + g * CPG + c) * H_ + h) * W_ + lane * 4;
    __builtin_amdgcn_global_load_async_to_lds_b128(
        glob_cast(ref + gofs), lds_cast(&Rl[wv][c][lane * 4]), 0, 0);
    __builtin_amdgcn_global_load_async_to_lds_b128(
        glob_cast(tgt + gofs), lds_cast(&Tl[wv][c][lane * 4]), 0, 0);
  }

  // --- wave32 WMMA fragment coordinates (ISA 7.12.2: VGPR v, half h -> K = v + 2h)
  const int half = lane >> 4;
  const int m16  = lane & 15;
  const int k0   = half * 2;

  // combined per-channel scale folded into A fragments (includes 1/8 mean):
  //   mean_c (R*sr)(T*st) = sum_c R * T * (sr*st/8)
  const float* mref = mask + b * C_ + g * CPG;
  const float* mtgt = mask + B_ * C_ + b * C_ + g * CPG;
  const int ks[4] = {k0, k0 + 1, k0 + 4, k0 + 5};
  float scl[4];
  #pragma unroll
  for (int q = 0; q < 4; ++q) {
    float sr = mref[ks[q]], st = mtgt[ks[q]];
    if (variant) { sr = 1.f - sr; st = 1.f - st; }
    scl[q] = sr * st * 0.125f;
  }

  asm volatile("s_wait_asynccnt 0x0" ::: "memory");  // DMA done before ds reads

  const int  t = 8 * half - m16;                     // d at r=0 for delta=0
  const long obase =
      ((long)((b * 2 + variant) * G_ + g) * D_) * (H_ * W_) + (long)h * W_;

  #pragma unroll
  for (int i = 0; i < 8; ++i) {                      // row tile: w in [16i,16i+16)
    const int wA = 16 * i + m16;
    v2f aLo, aHi;
    aLo.x = Rl[wv][k0 + 0][wA] * scl[0];  aLo.y = Rl[wv][k0 + 1][wA] * scl[1];
    aHi.x = Rl[wv][k0 + 4][wA] * scl[2];  aHi.y = Rl[wv][k0 + 5][wA] * scl[3];
    const int dmax = i < 3 ? i : 3;
    #pragma unroll
    for (int delta = 0; delta <= dmax; ++delta) {    // band: 0 <= i-j <= 3
      const int j  = i - delta;
      const int wB = 16 * j + m16;
      v2f bLo, bHi;
      bLo.x = Tl[wv][k0 + 0][wB];  bLo.y = Tl[wv][k0 + 1][wB];
      bHi.x = Tl[wv][k0 + 4][wB];  bHi.y = Tl[wv][k0 + 5][wB];
      v8f acc = {};
      acc = __builtin_amdgcn_wmma_f32_16x16x4_f32(false, aLo, false, bLo,
                                                  (short)0, acc, false, false);
      acc = __builtin_amdgcn_wmma_f32_16x16x4_f32(false, aHi, false, bHi,
                                                  (short)0, acc, false, false);
      // D row M = r + 8*half, col N = m16; consecutive r -> addr stride 8193
      float* p = out + obase + (long)(16 * delta + t) * (H_ * W_)
                     + (16 * i + 8 * half);
      if (delta == 0) {                              // keep d >= 0
        #pragma unroll
        for (int r = 0; r < 8; ++r)
          if (r + t >= 0) p[r * (H_ * W_ + 1)] = acc[r];
      } else if (delta == 3) {                       // keep d < 48
        #pragma unroll
        for (int r = 0; r < 8; ++r)
          if (r + t < 0) p[r * (H_ * W_ + 1)] = acc[r];
      } else {                                       // delta 1,2: whole tile in band
        #pragma unroll
        for (int r = 0; r < 8; ++r)
          p[r * (H_ * W_ + 1)] = acc[r];
      }
    }
  }
}

extern "C" void kernel_launch(void* const* d_in, const int* in_sizes, int n_in,
                              void* d_out, int out_size, void* d_ws, size_t ws_size,
                              hipStream_t stream) {
  const float* ref = (const float*)d_in[0];
  const float* tgt = (const float*)d_in[1];
  const float* w1  = (const float*)d_in[2];
  const float* b1  = (const float*)d_in[3];
  const float* w2  = (const float*)d_in[4];
  const float* b2  = (const float*)d_in[5];
  float* out  = (float*)d_out;
  float* gap  = (float*)d_ws;            // 2*B*C floats
  float* mskb = gap + 2 * B_ * C_;       // 2*B*C floats

  gap_kernel <<<2 * B_ * C_, 256, 0, stream>>>(ref, tgt, gap);
  mask_kernel<<<2 * B_,       64, 0, stream>>>(gap, w1, b1, w2, b2, mskb);
  ztri_kernel<<<B_ * 2 * G_ * D_, 64, 0, stream>>>(out);
  gwc_wmma_kernel<<<(B_ * 2 * H_ * G_) / 4, 128, 0, stream>>>(ref, tgt, mskb, out);
}